// GATEncoder_18562848653421
// MI455X (gfx1250) — compile-verified
//
#include <hip/hip_runtime.h>

typedef __attribute__((ext_vector_type(2))) float v2f;
typedef __attribute__((ext_vector_type(8))) float v8f;

#define NEG_SLOPE 0.2f
#define NEG_INF_F (-1e30f)

// ---- order-preserving float <-> uint mapping for atomic segment-max ----
__device__ __forceinline__ unsigned f2o(float f) {
    unsigned u = __float_as_uint(f);
    return (u & 0x80000000u) ? ~u : (u | 0x80000000u);
}
__device__ __forceinline__ float o2f(unsigned o) {
    return (o & 0x80000000u) ? __uint_as_float(o & 0x7FFFFFFFu)
                             : __uint_as_float(~o);
}

// =====================================================================
// GEMM via fp32 WMMA (16x16x4), one wave computes a 16-row strip across
// all COUT columns; fused epilogue produces s = h.a_src, d = h.a_dst.
// =====================================================================
template<int CIN, int COUT>
__global__ __launch_bounds__(256)
void gat_gemm_wmma(const float* __restrict__ X, const float* __restrict__ Wm,
                   const float* __restrict__ asrc, const float* __restrict__ adst,
                   float* __restrict__ H, float* __restrict__ S, float* __restrict__ D,
                   int N)
{
    constexpr int NT = COUT / 16;
    const int wave = (blockIdx.x * blockDim.x + threadIdx.x) >> 5;
    const int lane = threadIdx.x & 31;
    const int m0 = wave * 16;
    if (m0 >= N) return;                    // wave-uniform: EXEC stays all-1 for WMMA

    const int half = lane >> 4;             // 0: lanes 0-15, 1: lanes 16-31
    const int l = lane & 15;

    v8f acc[NT] = {};

    // A fragment: lane (half,l) holds X[m0+l][k+2*half .. k+2*half+1]
    int arow = m0 + l; if (arow >= N) arow = N - 1;
    const float* xrow = X + (size_t)arow * CIN + 2 * half;

    #pragma unroll 4
    for (int k = 0; k < CIN; k += 4) {
        v2f A;
        A[0] = xrow[k];
        A[1] = xrow[k + 1];
        #pragma unroll
        for (int t = 0; t < NT; ++t) {
            const int n = t * 16 + l;
            v2f B;   // B fragment: lane (half,l) holds W[k+2*half + {0,1}][n]
            B[0] = Wm[(size_t)(k + 2 * half    ) * COUT + n];
            B[1] = Wm[(size_t)(k + 2 * half + 1) * COUT + n];
            acc[t] = __builtin_amdgcn_wmma_f32_16x16x4_f32(
                         false, A, false, B, (short)0, acc[t], false, false);
        }
    }

    // C/D layout: VGPR j, lanes 0-15 -> row m0+j, lanes 16-31 -> row m0+8+j
    #pragma unroll
    for (int j = 0; j < 8; ++j) {
        const int row = m0 + j + 8 * half;
        float ps = 0.0f, pd = 0.0f;
        #pragma unroll
        for (int t = 0; t < NT; ++t) {
            const int n = t * 16 + l;
            const float v = acc[t][j];
            if (row < N) H[(size_t)row * COUT + n] = v;
            ps += v * asrc[n];
            pd += v * adst[n];
        }
        // reduce over the 16 lanes of this half (xor<16 never crosses halves)
        #pragma unroll
        for (int off = 1; off < 16; off <<= 1) {
            ps += __shfl_xor(ps, off, 32);
            pd += __shfl_xor(pd, off, 32);
        }
        if (l == 0 && row < N) { S[row] = ps; D[row] = pd; }
    }
}

// =====================================================================
// Per-layer scratch init (graph-replay safe: runs every call)
// =====================================================================
__global__ void gat_init(float* __restrict__ agg, unsigned* __restrict__ Mo,
                         float* __restrict__ den, int N, int COUT)
{
    const int i = blockIdx.x * blockDim.x + threadIdx.x;
    if (i < N * COUT) agg[i] = 0.0f;
    if (i < N) { Mo[i] = 0u; den[i] = 0.0f; }   // f2o-encoded 0u == "-inf"
}

// =====================================================================
// Edge pass 1: alpha = leaky_relu(s[src]+d[dst]); mask original self edges;
// segment-max via ordered-uint atomicMax.
// =====================================================================
__global__ void gat_edge_alpha(const int* __restrict__ src, const int* __restrict__ dst,
                               const float* __restrict__ S, const float* __restrict__ D,
                               float* __restrict__ alp, unsigned* __restrict__ Mo,
                               int E, int Eall)
{
    const int e = blockIdx.x * blockDim.x + threadIdx.x;
    if (e >= Eall) return;
    int si, di; bool self;
    if (e < E) { si = src[e]; di = dst[e]; self = (si == di); }
    else       { si = di = e - E;          self = false; }
    float a = S[si] + D[di];
    a = (a > 0.0f) ? a : NEG_SLOPE * a;
    if (self) a = NEG_INF_F;
    alp[e] = a;
    atomicMax(&Mo[di], f2o(a));
}

// =====================================================================
// Edge pass 2: e = exp(alpha - max[dst]) (stored in place); denom[dst] += e
// =====================================================================
__global__ void gat_edge_exp(const int* __restrict__ dst,
                             float* __restrict__ alp, const unsigned* __restrict__ Mo,
                             float* __restrict__ den, int E, int Eall)
{
    const int e = blockIdx.x * blockDim.x + threadIdx.x;
    if (e >= Eall) return;
    const int di = (e < E) ? dst[e] : (e - E);
    const float m = o2f(Mo[di]);
    const float ex = __expf(alp[e] - m);   // masked edges: exp(-1e30-m) -> 0
    alp[e] = ex;
    atomicAdd(&den[di], ex);
}

// =====================================================================
// Edge pass 3 (heavy): agg[dst] += (e/denom[dst]) * h[src]
// COUT/4 lanes per edge, float4 gathers, f32 atomics into L2.
// =====================================================================
template<int COUT>
__global__ __launch_bounds__(256)
void gat_edge_agg(const int* __restrict__ src, const int* __restrict__ dst,
                  const float* __restrict__ ex, const float* __restrict__ den,
                  const float* __restrict__ H, float* __restrict__ AGG,
                  int E, int Eall)
{
    constexpr int G = COUT / 4;
    const long long tid = (long long)blockIdx.x * blockDim.x + threadIdx.x;
    const long long e = tid / G;
    if (e >= Eall) return;
    const int c4 = (int)(tid % G) * 4;
    int si, di;
    if (e < E) { si = src[e]; di = dst[e]; }
    else       { si = di = (int)(e - E); }
    const float w = ex[e] / den[di];
    const float4 hv = *(const float4*)(H + (size_t)si * COUT + c4);
    float* outp = AGG + (size_t)di * COUT + c4;
    atomicAdd(outp + 0, w * hv.x);
    atomicAdd(outp + 1, w * hv.y);
    atomicAdd(outp + 2, w * hv.z);
    atomicAdd(outp + 3, w * hv.w);
}

// =====================================================================
// Bias + optional ReLU
// =====================================================================
template<bool RELU, int COUT>
__global__ void gat_finalize(const float* __restrict__ agg, const float* __restrict__ b,
                             float* __restrict__ out, int N)
{
    const int i = blockIdx.x * blockDim.x + threadIdx.x;
    if (i >= N * COUT) return;
    float v = agg[i] + b[i & (COUT - 1)];
    if (RELU) v = fmaxf(v, 0.0f);
    out[i] = v;
}

static inline int cdiv_ll(long long a, long long b) { return (int)((a + b - 1) / b); }

extern "C" void kernel_launch(void* const* d_in, const int* in_sizes, int n_in,
                              void* d_out, int out_size, void* d_ws, size_t ws_size,
                              hipStream_t stream)
{
    const float* x   = (const float*)d_in[0];
    const int*   ei  = (const int*)  d_in[1];
    const float* W1  = (const float*)d_in[2];
    const float* as1 = (const float*)d_in[3];
    const float* ad1 = (const float*)d_in[4];
    const float* b1  = (const float*)d_in[5];
    const float* W2  = (const float*)d_in[6];
    const float* as2 = (const float*)d_in[7];
    const float* ad2 = (const float*)d_in[8];
    const float* b2  = (const float*)d_in[9];
    const float* W3  = (const float*)d_in[10];
    const float* as3 = (const float*)d_in[11];
    const float* ad3 = (const float*)d_in[12];
    const float* b3  = (const float*)d_in[13];

    const int N    = in_sizes[0] / 128;
    const int E    = in_sizes[1] / 2;
    const int Eall = E + N;
    const int* src = ei;
    const int* dst = ei + E;

    // ---- workspace layout (floats) ----
    float* ws   = (float*)d_ws;
    float* h    = ws;                          // N*128 (reused smaller in L2/L3)
    float* agg  = h   + (size_t)N * 128;       // N*128
    float* xn   = agg + (size_t)N * 128;       // N*128
    float* alp  = xn  + (size_t)N * 128;       // Eall
    float* Sv   = alp + Eall;                  // N
    float* Dv   = Sv  + N;                     // N
    unsigned* Mo = (unsigned*)(Dv + N);        // N
    float* den  = (float*)(Mo + N);            // N

    const dim3 blk(256);
    const int waves_blocks = cdiv_ll((long long)cdiv_ll(N, 16) * 32, 256);
    const int edge_blocks  = cdiv_ll(Eall, 256);

    // ================= Layer 1: 128 -> 128, ReLU =================
    gat_init<<<cdiv_ll((long long)N * 128, 256), blk, 0, stream>>>(agg, Mo, den, N, 128);
    gat_gemm_wmma<128,128><<<waves_blocks, blk, 0, stream>>>(x, W1, as1, ad1, h, Sv, Dv, N);
    gat_edge_alpha<<<edge_blocks, blk, 0, stream>>>(src, dst, Sv, Dv, alp, Mo, E, Eall);
    gat_edge_exp<<<edge_blocks, blk, 0, stream>>>(dst, alp, Mo, den, E, Eall);
    gat_edge_agg<128><<<cdiv_ll((long long)Eall * 32, 256), blk, 0, stream>>>(src, dst, alp, den, h, agg, E, Eall);
    gat_finalize<true,128><<<cdiv_ll((long long)N * 128, 256), blk, 0, stream>>>(agg, b1, xn, N);

    // ================= Layer 2: 128 -> 64, ReLU =================
    gat_init<<<cdiv_ll((long long)N * 64, 256), blk, 0, stream>>>(agg, Mo, den, N, 64);
    gat_gemm_wmma<128,64><<<waves_blocks, blk, 0, stream>>>(xn, W2, as2, ad2, h, Sv, Dv, N);
    gat_edge_alpha<<<edge_blocks, blk, 0, stream>>>(src, dst, Sv, Dv, alp, Mo, E, Eall);
    gat_edge_exp<<<edge_blocks, blk, 0, stream>>>(dst, alp, Mo, den, E, Eall);
    gat_edge_agg<64><<<cdiv_ll((long long)Eall * 16, 256), blk, 0, stream>>>(src, dst, alp, den, h, agg, E, Eall);
    gat_finalize<true,64><<<cdiv_ll((long long)N * 64, 256), blk, 0, stream>>>(agg, b2, xn, N);

    // ================= Layer 3: 64 -> 64, no activation, -> d_out =================
    gat_init<<<cdiv_ll((long long)N * 64, 256), blk, 0, stream>>>(agg, Mo, den, N, 64);
    gat_gemm_wmma<64,64><<<waves_blocks, blk, 0, stream>>>(xn, W3, as3, ad3, h, Sv, Dv, N);
    gat_edge_alpha<<<edge_blocks, blk, 0, stream>>>(src, dst, Sv, Dv, alp, Mo, E, Eall);
    gat_edge_exp<<<edge_blocks, blk, 0, stream>>>(dst, alp, Mo, den, E, Eall);
    gat_edge_agg<64><<<cdiv_ll((long long)Eall * 16, 256), blk, 0, stream>>>(src, dst, alp, den, h, agg, E, Eall);
    gat_finalize<false,64><<<cdiv_ll((long long)N * 64, 256), blk, 0, stream>>>(agg, b3, (float*)d_out, N);
}